// HistogramLoss_57664230916310
// MI455X (gfx1250) — compile-verified
//
#include <hip/hip_runtime.h>

// HistogramLoss pipeline for gfx1250 (MI455X), compile-only target.
// Structure: init -> fused elementwise -> LDS-histogram gather -> WMMA cumsum
// CDF + transfer table (single WGP) -> scatter matched values -> L1 mean.

#define H 1024
#define HH (H * H)
#define CHW (3 * HH)
#define NBINS 256
#define LOSS_OFF (4 * CHW)

typedef __attribute__((ext_vector_type(2))) float v2f;
typedef __attribute__((ext_vector_type(8))) float v8f;

__device__ __forceinline__ int iclamp(int v, int lo, int hi) {
  return v < lo ? lo : (v > hi ? hi : v);
}

// de_norm(x)*255 exactly as reference: clip((x+1)/2, 0, 1) * 255
__device__ __forceinline__ float denorm255(float x) {
  float d = (x + 1.0f) / 2.0f;
  d = fminf(fmaxf(d, 0.0f), 1.0f);
  return d * 255.0f;
}

// ---------------------------------------------------------------- init ------
__global__ void init_kernel(unsigned* __restrict__ hist_g, float* __restrict__ loss) {
  for (int i = threadIdx.x; i < 6 * NBINS; i += 256) hist_g[i] = 0u;
  if (threadIdx.x == 0) *loss = 0.0f;
}

// --------------------------------------------------------- elementwise ------
// Writes out0=target_masked/255, out1=ref_masked/255, out2=input_masked/255,
// out3 initialized to ref_masked/255 (scatter overwrites matched positions).
__global__ void elementwise_kernel(const float* __restrict__ input,
                                   const float* __restrict__ target,
                                   const float* __restrict__ ref,
                                   const float* __restrict__ msrc,
                                   const float* __restrict__ mtar,
                                   float* __restrict__ out) {
  int t = blockIdx.x * blockDim.x + threadIdx.x;
  int base = t * 4;                 // 4 consecutive elems, never crosses channel
  if (base >= CHW) return;
  int p = base % HH;                // pixel index within channel (mask index)

  float4 tv = *(const float4*)(target + base);
  float4 iv = *(const float4*)(input + base);
  float4 rv = *(const float4*)(ref + base);
  float4 msv = *(const float4*)(msrc + p);
  float4 mtv = *(const float4*)(mtar + p);

  const float* tp = &tv.x;  const float* ip = &iv.x;  const float* rp = &rv.x;
  const float* mp = &msv.x; const float* np = &mtv.x;
  float4 o0, o1, o2;
  float* o0p = &o0.x; float* o1p = &o1.x; float* o2p = &o2.x;

#pragma unroll
  for (int k = 0; k < 4; ++k) {
    float ms = mp[k] / 255.0f;
    float mt = np[k] / 255.0f;
    float tm = denorm255(tp[k]) * mt;                             // target_masked
    float rm = denorm255(rp[k]) * ms;                             // ref_masked
    float im = fminf(fmaxf(ip[k], 0.0f), 1.0f) * 255.0f * ms;     // input_masked
    o0p[k] = tm / 255.0f;
    o1p[k] = rm / 255.0f;
    o2p[k] = im / 255.0f;
  }
  *(float4*)(out + base)           = o0;
  *(float4*)(out + CHW + base)     = o1;
  *(float4*)(out + 2 * CHW + base) = o2;
  *(float4*)(out + 3 * CHW + base) = o1;   // input_match base = ref_masked/255
}

// -------------------------------------------------------- gather + hist -----
// hist_g layout: [t*256+b], t=0..2 dst(ch), t=3..5 ref(ch)
__global__ void gather_hist_kernel(const float* __restrict__ target,
                                   const float* __restrict__ ref,
                                   const float* __restrict__ msrc,
                                   const float* __restrict__ mtar,
                                   const int* __restrict__ i0, const int* __restrict__ i1,
                                   const int* __restrict__ i2, const int* __restrict__ i3,
                                   unsigned* __restrict__ hist_g, int n) {
  __shared__ unsigned lh[6 * NBINS];
  for (int i = threadIdx.x; i < 6 * NBINS; i += 256) lh[i] = 0u;
  __syncthreads();

  int i = blockIdx.x * 256 + threadIdx.x;
  if (i < n) {
    int pd = i0[i] * H + i1[i];
    int pr = i2[i] * H + i3[i];
    float msd = msrc[pd] / 255.0f;
    float mtr = mtar[pr] / 255.0f;
#pragma unroll
    for (int c = 0; c < 3; ++c) {
      float dv = denorm255(ref[c * HH + pd]) * msd;      // ref_masked sample
      atomicAdd(&lh[c * NBINS + iclamp((int)dv, 0, NBINS - 1)], 1u);
      float rv2 = denorm255(target[c * HH + pr]) * mtr;  // target_masked sample
      atomicAdd(&lh[(3 + c) * NBINS + iclamp((int)rv2, 0, NBINS - 1)], 1u);
    }
  }
  __syncthreads();
  for (int j = threadIdx.x; j < 6 * NBINS; j += 256) {
    unsigned v = lh[j];
    if (v) atomicAdd(&hist_g[j], v);
  }
}

// ----------------------------------------------- CDF (WMMA cumsum) + table --
// One workgroup of 256 threads = 8 waves. Waves 0..5 each own one 256-bin
// cumsum. hist viewed as 16x16 Hm; S = Hm x U (upper-tri ones) via four
// chained V_WMMA_F32_16X16X4_F32 (K in chunks of 4), then scalar row-prefix +
// normalize. Then all 256 threads build the transfer tables.
__global__ void cdf_table_kernel(const unsigned* __restrict__ hist_g,
                                 int* __restrict__ tables_g) {
  __shared__ float h[6 * NBINS];
  __shared__ float cdf[6 * NBINS];
  int tid = threadIdx.x;

  for (int i = tid; i < 6 * NBINS; i += 256) h[i] = (float)hist_g[i];
  __syncthreads();

  int wave = tid >> 5;
  int lane = tid & 31;
  if (wave < 6) {
    const float* hh = &h[wave * NBINS];
    float* cc = &cdf[wave * NBINS];
    int m = lane & 15;        // A-row / B-col / D-col for this lane
    int khalf = lane >> 4;    // lanes 16-31 carry the upper K pair / rows +8

#if defined(__AMDGCN__) && __has_builtin(__builtin_amdgcn_wmma_f32_16x16x4_f32)
    v8f acc = {};
#pragma unroll
    for (int c = 0; c < 4; ++c) {
      int k0 = 4 * c + 2 * khalf;       // this lane's K pair within chunk c
      v2f a, b;
      a.x = hh[m * 16 + k0];            // A(m, k0)
      a.y = hh[m * 16 + k0 + 1];        // A(m, k0+1)
      b.x = (k0 <= m) ? 1.0f : 0.0f;    // U(k0, n), n == lane&15 == m
      b.y = (k0 + 1 <= m) ? 1.0f : 0.0f;
      acc = __builtin_amdgcn_wmma_f32_16x16x4_f32(
          /*neg_a=*/false, a, /*neg_b=*/false, b,
          /*c_mod=*/(short)0, acc, /*reuse_a=*/false, /*reuse_b=*/false);
    }
    // D layout: vgpr v -> row v + 8*khalf, col = lane&15
#pragma unroll
    for (int v = 0; v < 8; ++v) cc[(v + 8 * khalf) * 16 + m] = acc[v];
#else
    // Scalar fallback (host pass / missing builtin): within-row inclusive scan
    if (lane == 0) {
      for (int r = 0; r < 16; ++r) {
        float s = 0.0f;
        for (int j = 0; j < 16; ++j) { s += hh[r * 16 + j]; cc[r * 16 + j] = s; }
      }
    }
#endif
    // Row prefix across the 16 rows (all lanes read row totals, then each
    // lane finalizes its 8 entries; wave-internal DS ops are in order).
    float rowpref[16];
    float run = 0.0f;
#pragma unroll
    for (int r = 0; r < 16; ++r) { rowpref[r] = run; run += cc[r * 16 + 15]; }
    float total = (run > 0.0f) ? run : 1.0f;
#pragma unroll
    for (int v = 0; v < 8; ++v) {
      int row = v + 8 * khalf;
      cc[row * 16 + m] = (cc[row * 16 + m] + rowpref[row]) / total;
    }
  }
  __syncthreads();

  // Transfer tables: table[i] = first j in [0,254] with
  // cdf_ref[j] <= cdf_dst[i] <= cdf_ref[j+1], then j+1; else i; table[255]=255.
  for (int ch = 0; ch < 3; ++ch) {
    float r = cdf[ch * NBINS + tid];
    const float* cr = &cdf[(3 + ch) * NBINS];
    int tab = tid;
    for (int j = 0; j < NBINS - 1; ++j) {
      if (cr[j] <= r && r <= cr[j + 1]) { tab = j + 1; break; }
    }
    if (tid == NBINS - 1) tab = NBINS - 1;
    tables_g[ch * NBINS + tid] = tab;
  }
}

// ------------------------------------------------------------- scatter ------
__global__ void scatter_kernel(const float* __restrict__ ref,
                               const float* __restrict__ msrc,
                               const int* __restrict__ i0, const int* __restrict__ i1,
                               const int* __restrict__ tables,
                               float* __restrict__ out3, int n) {
  int i = blockIdx.x * 256 + threadIdx.x;
  if (i >= n) return;
  int pd = i0[i] * H + i1[i];
  float msd = msrc[pd] / 255.0f;
#pragma unroll
  for (int c = 0; c < 3; ++c) {
    float dv = denorm255(ref[c * HH + pd]) * msd;
    int bd = iclamp((int)dv, 0, NBINS - 1);
    out3[c * HH + pd] = (float)tables[c * NBINS + bd] / 255.0f;  // benign dup race
  }
}

// ---------------------------------------------------------------- loss ------
__global__ void loss_kernel(const float* __restrict__ a, const float* __restrict__ b,
                            float* __restrict__ loss, int n4) {
  float s = 0.0f;
  for (int i = blockIdx.x * blockDim.x + threadIdx.x; i < n4;
       i += gridDim.x * blockDim.x) {
    float4 x = ((const float4*)a)[i];
    float4 y = ((const float4*)b)[i];
    s += fabsf(x.x - y.x) + fabsf(x.y - y.y) + fabsf(x.z - y.z) + fabsf(x.w - y.w);
  }
  __shared__ float red[256];
  red[threadIdx.x] = s;
  __syncthreads();
  for (int off = 128; off > 0; off >>= 1) {
    if (threadIdx.x < off) red[threadIdx.x] += red[threadIdx.x + off];
    __syncthreads();
  }
  if (threadIdx.x == 0) atomicAdd(loss, red[0] * (1.0f / (float)CHW));
}

// ---------------------------------------------------------------------------
extern "C" void kernel_launch(void* const* d_in, const int* in_sizes, int n_in,
                              void* d_out, int out_size, void* d_ws, size_t ws_size,
                              hipStream_t stream) {
  const float* input  = (const float*)d_in[0];
  const float* target = (const float*)d_in[1];
  const float* ref    = (const float*)d_in[2];
  const float* msrc   = (const float*)d_in[3];
  const float* mtar   = (const float*)d_in[4];
  // d_in[5] = target_data_eye (unused by reference)
  const int* i0 = (const int*)d_in[6];
  const int* i1 = (const int*)d_in[7];
  const int* i2 = (const int*)d_in[8];
  const int* i3 = (const int*)d_in[9];
  int n = in_sizes[6];

  float* out = (float*)d_out;
  unsigned* hist_g = (unsigned*)d_ws;                                  // 6*256 u32
  int* tables_g = (int*)((char*)d_ws + 6 * NBINS * sizeof(unsigned));  // 3*256 i32

  init_kernel<<<1, 256, 0, stream>>>(hist_g, out + LOSS_OFF);
  elementwise_kernel<<<CHW / 4 / 256, 256, 0, stream>>>(input, target, ref,
                                                        msrc, mtar, out);
  int gb = (n + 255) / 256;
  gather_hist_kernel<<<gb, 256, 0, stream>>>(target, ref, msrc, mtar,
                                             i0, i1, i2, i3, hist_g, n);
  cdf_table_kernel<<<1, 256, 0, stream>>>(hist_g, tables_g);
  scatter_kernel<<<gb, 256, 0, stream>>>(ref, msrc, i0, i1, tables_g,
                                         out + 3 * CHW, n);
  loss_kernel<<<1024, 256, 0, stream>>>(out + 2 * CHW, out + 3 * CHW,
                                        out + LOSS_OFF, CHW / 4);
}